// Conduits_20486994002256
// MI455X (gfx1250) — compile-verified
//
#include <hip/hip_runtime.h>
#include <stdint.h>

// ---------------------------------------------------------------------------
// Problem constants (raster mesh is structurally fixed by the reference).
// ---------------------------------------------------------------------------
constexpr int   NRg  = 1024, NCg = 1024;
constexpr int   NHl  = NRg * (NCg - 1);            // # horizontal links
constexpr float Gc   = 9.81f;
constexpr float RHOI = 917.0f;
constexpr float RHOW = 1000.0f;
constexpr float NUc  = 1.787e-6f;
constexpr float OMG  = 1e-3f;
constexpr float LHc  = 334000.0f;
constexpr float AFLUc= 6e-24f;

#define DEV __device__ __forceinline__

// ---------------------------------------------------------------------------
// CDNA5 feature probes (compile-only loop: guard everything).
// ---------------------------------------------------------------------------
#if defined(__AMDGCN__)
#  if __has_builtin(__builtin_amdgcn_global_load_async_to_lds_b32)
#    define HAVE_ASYNC_LDS 1
#  endif
#  if __has_builtin(__builtin_amdgcn_wmma_f32_16x16x4_f32)
#    define HAVE_WMMA_F32 1
#  endif
#endif

typedef float v2f __attribute__((ext_vector_type(2)));
typedef float v8f __attribute__((ext_vector_type(8)));

#if defined(__AMDGCN__)
typedef __attribute__((address_space(1))) int as1_int;
typedef __attribute__((address_space(3))) int as3_int;
#endif

DEV bool is_bdry(int r, int c) {
  return (r == 0) | (r == NRg - 1) | (c == 0) | (c == NCg - 1);
}

// Global -> LDS staged copy. On gfx1250 this lowers to
// global_load_async_to_lds_b32 (ASYNCcnt-tracked); else a plain ld/st pair.
DEV void g2s_f32(float* dst_lds, const float* src_glb) {
#if defined(HAVE_ASYNC_LDS)
  __builtin_amdgcn_global_load_async_to_lds_b32(
      (as1_int*)(src_glb), (as3_int*)(dst_lds), 0, 0);
#else
  *dst_lds = *src_glb;
#endif
}

DEV void async_wait_all() {
#if defined(HAVE_ASYNC_LDS)
#  if __has_builtin(__builtin_amdgcn_s_wait_asynccnt)
  __builtin_amdgcn_s_wait_asynccnt(0);
#  else
  asm volatile("s_wait_asynccnt 0x0" ::: "memory");
#  endif
#endif
}

// 32-lane (wave32) sum. WMMA trick: A(16x4) holds lane values (K=0 for lanes
// 0-15, K=2 for lanes 16-31), B = ones => D[m,j] = v[m]+v[m+16] for every j.
// Lane j sums its 8 accumulator VGPRs (rows 0-7 or 8-15), one shfl_xor(16)
// combines the halves. EXEC must be all ones at every call site (it is).
DEV float wave_sum32(float v) {
#if defined(HAVE_WMMA_F32)
  v2f a; a.x = v;    a.y = 0.0f;
  v2f b; b.x = 1.0f; b.y = 1.0f;
  v8f c = {};
  c = __builtin_amdgcn_wmma_f32_16x16x4_f32(false, a, false, b,
                                            (short)0, c, false, false);
  float s = 0.0f;
#pragma unroll
  for (int i = 0; i < 8; ++i) s += c[i];
  s += __shfl_xor(s, 16, 32);
  return s;
#else
#pragma unroll
  for (int off = 16; off; off >>= 1) v += __shfl_xor(v, off, 32);
  return v;
#endif
}

// Deterministic 256-thread block reduction -> one partial per block.
DEV void block_accum256(float v, float* red8, float* out_partial, int tid) {
  float w = wave_sum32(v);
  if ((tid & 31) == 0) red8[tid >> 5] = w;
  __syncthreads();
  if (tid == 0) {
    float s = 0.0f;
#pragma unroll
    for (int i = 0; i < 8; ++i) s += red8[i];
    *out_partial = s;
  }
}

// ---------------------------------------------------------------------------
// Kernel 0: x0 = hydraulic_head (CG solution lives directly in d_out[N:2N]).
// ---------------------------------------------------------------------------
__global__ __launch_bounds__(256) void copy_kernel(const float* __restrict__ a,
                                                   float* __restrict__ o, int n) {
  int i = blockIdx.x * 256 + threadIdx.x;
  if (i < n) o[i] = a[i];
}

// ---------------------------------------------------------------------------
// Kernel 1: per-link Picard iteration -> Re (output), T, melt_links.
// Link ids are structural: [0,NHl) horizontal row-major, rest vertical.
// ---------------------------------------------------------------------------
__global__ __launch_bounds__(256) void link_kernel(
    const float* __restrict__ S, const float* __restrict__ hh,
    const float* __restrict__ rey, const float* __restrict__ llen,
    float* __restrict__ T, float* __restrict__ ML, float* __restrict__ outRe,
    int L) {
  int l = blockIdx.x * 256 + threadIdx.x;
  if (l >= L) return;
  int tail, head;
  if (l < NHl) {
    int rr = l / (NCg - 1), cc = l - rr * (NCg - 1);
    tail = rr * NCg + cc;  head = tail + 1;
  } else {
    int j = l - NHl;           // j == r*NCg + c for vertical links
    tail = j;  head = j + NCg;
  }
  float Sl = 0.5f * (S[head] + S[tail]);
  float gh = (hh[head] - hh[tail]) / llen[l];
  float K  = Sl * Sl * Sl * (Gc / (12.0f * NUc));
  float A0 = fabsf(K * gh) * (1.0f / NUc);   // |T(Re)*gh|/nu = A0/(1+w*Re)
  float Re = rey[l];
#pragma unroll
  for (int it = 0; it < 15; ++it) Re = A0 / (1.0f + OMG * Re);
  outRe[l] = Re;
  float Tl = K / (1.0f + OMG * Re);
  T[l]  = Tl;
  ML[l] = (RHOW * Gc) * fabsf(Tl * gh * gh); // rho_w*g*|Q*gh|, Q=T*gh
}

// ---------------------------------------------------------------------------
// Kernel 2: per-node melt/forcing + RK4 conduit update (-1 wraps to link L-1,
// faithfully reproducing JAX negative indexing).
// ---------------------------------------------------------------------------
__global__ __launch_bounds__(256) void node_kernel(
    const float* __restrict__ S, const float* __restrict__ hh,
    const float* __restrict__ H, const float* __restrict__ bed,
    const float* __restrict__ mw, const float* __restrict__ ghf,
    const float* __restrict__ ML, const int* __restrict__ dtp,
    float* __restrict__ F, float* __restrict__ newS, int n, int L) {
  int i = blockIdx.x * 256 + threadIdx.x;
  if (i >= n) return;
  int r = i / NCg, c = i - r * NCg;
  int le = (c < NCg - 1) ? r * (NCg - 1) + c     : L - 1;
  int lw = (c > 0)       ? r * (NCg - 1) + c - 1 : L - 1;
  int ln = (r < NRg - 1) ? NHl + i               : L - 1;
  int ls = (r > 0)       ? NHl + i - NCg         : L - 1;
  float mn = 0.25f * (ML[le] + ML[ln] + ML[lw] + ML[ls]);
  float melt_rate = (ghf[i] + mn) * (1.0f / LHc);
  float Neff = RHOI * Gc * H[i] - RHOW * Gc * (hh[i] - bed[i]);
  float melt_term = melt_rate * (1.0f / RHOW - 1.0f / RHOI);
  float closure = AFLUc * Neff * Neff * Neff * S[i];
  F[i] = melt_term + closure + mw[i];
  float m  = melt_term * (1.0f / RHOI);
  float cc = closure;
  float dtf = (float)dtp[0];
  float S0 = S[i];
  float k1 = m - cc * S0;
  float k2 = m - cc * (S0 + 0.5f * dtf * k1);
  float k3 = m - cc * (S0 + 0.5f * dtf * k2);
  float k4 = m - cc * (S0 + dtf * k3);
  newS[i] = S0 + dtf * (k1 + 2.0f * k2 + 2.0f * k3 + k4) * (1.0f / 6.0f);
}

// ---------------------------------------------------------------------------
// Kernel 3: b = matvec_T(forcing) (stand-alone 5-point stencil; run once).
// vt = +Te(yiC+yiE) + Tn(yiC+yiN) - Tw(yiW+yiC) - Ts(yiS+yiC); out = vt + 1b*y
// ---------------------------------------------------------------------------
__global__ __launch_bounds__(256) void matvecT_kernel(
    const float* __restrict__ y, const float* __restrict__ T,
    const float* __restrict__ area, float* __restrict__ out, int n) {
  int i = blockIdx.x * 256 + threadIdx.x;
  if (i >= n) return;
  int r = i / NCg, c = i - r * NCg;
  bool isbC = is_bdry(r, c);
  float yiC = isbC ? 0.0f : y[i] / area[i];
  float vt = 0.0f;
  if (c < NCg - 1) {
    int j = i + 1;
    float yiE = is_bdry(r, c + 1) ? 0.0f : y[j] / area[j];
    vt += T[r * (NCg - 1) + c] * (yiC + yiE);
  }
  if (r < NRg - 1) {
    int j = i + NCg;
    float yiN = is_bdry(r + 1, c) ? 0.0f : y[j] / area[j];
    vt += T[NHl + i] * (yiC + yiN);
  }
  if (c > 0) {
    int j = i - 1;
    float yiW = is_bdry(r, c - 1) ? 0.0f : y[j] / area[j];
    vt -= T[r * (NCg - 1) + c - 1] * (yiW + yiC);
  }
  if (r > 0) {
    int j = i - NCg;
    float yiS = is_bdry(r - 1, c) ? 0.0f : y[j] / area[j];
    vt -= T[NHl + j] * (yiS + yiC);
  }
  out[i] = vt + (isbC ? y[i] : 0.0f);
}

// ---------------------------------------------------------------------------
// Kernel 4 (hot loop body): Ap = A^T (A p), fused via LDS tiles.
//   sh_p : p with halo 2 (async-staged into LDS)
//   sh_t : tmp = A*p with halo 1 (computed in LDS)
// Also emits per-block partial of dot(p, Ap).
// ---------------------------------------------------------------------------
__global__ __launch_bounds__(256) void normal_op_kernel(
    const float* __restrict__ p, const float* __restrict__ T,
    const float* __restrict__ area, float* __restrict__ Ap,
    float* __restrict__ partials) {
  __shared__ float sh_p[20][21];
  __shared__ float sh_t[18][19];
  __shared__ float red8[8];
  const int tx = threadIdx.x, ty = threadIdx.y;
  const int tid = ty * 16 + tx;
  const int r0 = blockIdx.y * 16, c0 = blockIdx.x * 16;

  // Stage p tile (20x20 incl. halo 2); out-of-grid -> 0.
  for (int s = tid; s < 400; s += 256) {
    int lr = s / 20, lc = s - lr * 20;
    int gr = r0 - 2 + lr, gc = c0 - 2 + lc;
    if (gr >= 0 && gr < NRg && gc >= 0 && gc < NCg)
      g2s_f32(&sh_p[lr][lc], p + gr * NCg + gc);
    else
      sh_p[lr][lc] = 0.0f;
  }
  async_wait_all();
  __syncthreads();

  // tmp = matvec(p) on 18x18 (tile + halo 1); T/area served from L2.
  for (int s = tid; s < 324; s += 256) {
    int lr = s / 18, lc = s - lr * 18;
    int gr = r0 - 1 + lr, gc = c0 - 1 + lc;
    float t = 0.0f;
    if (gr >= 0 && gr < NRg && gc >= 0 && gc < NCg) {
      float pc = sh_p[lr + 1][lc + 1];
      if (is_bdry(gr, gc)) {
        t = pc;
      } else {
        float pe = sh_p[lr + 1][lc + 2], pw = sh_p[lr + 1][lc];
        float pn = sh_p[lr + 2][lc + 1], ps = sh_p[lr][lc + 1];
        float Te = T[gr * (NCg - 1) + gc], Tw = T[gr * (NCg - 1) + gc - 1];
        float Tn = T[NHl + gr * NCg + gc], Ts = T[NHl + (gr - 1) * NCg + gc];
        float sf = -Te * (pe - pc) - Tn * (pn - pc)
                   - Tw * (pc - pw) - Ts * (pc - ps);
        t = sf / area[gr * NCg + gc];
      }
    }
    sh_t[lr][lc] = t;
  }
  __syncthreads();

  // Ap = matvec_T(tmp) on the 16x16 interior.
  const int gr = r0 + ty, gc = c0 + tx;
  float yC = sh_t[ty + 1][tx + 1];
  bool isbC = is_bdry(gr, gc);
  float yiC = isbC ? 0.0f : yC / area[gr * NCg + gc];
  float vt = 0.0f;
  if (gc < NCg - 1) {
    float yiE = is_bdry(gr, gc + 1) ? 0.0f
              : sh_t[ty + 1][tx + 2] / area[gr * NCg + gc + 1];
    vt += T[gr * (NCg - 1) + gc] * (yiC + yiE);
  }
  if (gr < NRg - 1) {
    float yiN = is_bdry(gr + 1, gc) ? 0.0f
              : sh_t[ty + 2][tx + 1] / area[(gr + 1) * NCg + gc];
    vt += T[NHl + gr * NCg + gc] * (yiC + yiN);
  }
  if (gc > 0) {
    float yiW = is_bdry(gr, gc - 1) ? 0.0f
              : sh_t[ty + 1][tx] / area[gr * NCg + gc - 1];
    vt -= T[gr * (NCg - 1) + gc - 1] * (yiW + yiC);
  }
  if (gr > 0) {
    float yiS = is_bdry(gr - 1, gc) ? 0.0f
              : sh_t[ty][tx + 1] / area[(gr - 1) * NCg + gc];
    vt -= T[NHl + (gr - 1) * NCg + gc] * (yiS + yiC);
  }
  float out = vt + (isbC ? yC : 0.0f);
  Ap[gr * NCg + gc] = out;

  float acc = sh_p[ty + 2][tx + 2] * out;   // p(r,c) * Ap(r,c)
  block_accum256(acc, red8, &partials[blockIdx.y * gridDim.x + blockIdx.x], tid);
}

// ---------------------------------------------------------------------------
// Kernel 5: r = b - A^TA x0; p = r; partial rr.
// ---------------------------------------------------------------------------
__global__ __launch_bounds__(256) void residual_kernel(
    const float* __restrict__ b, const float* __restrict__ Ax,
    float* __restrict__ r, float* __restrict__ p,
    float* __restrict__ partials, int n) {
  __shared__ float red8[8];
  int i = blockIdx.x * 256 + threadIdx.x;
  float acc = 0.0f;
  if (i < n) {
    float rv = b[i] - Ax[i];
    r[i] = rv;  p[i] = rv;  acc = rv * rv;
  }
  block_accum256(acc, red8, &partials[blockIdx.x], threadIdx.x);
}

// ---------------------------------------------------------------------------
// Kernel 6: deterministic single-block reduction of block partials -> scalar.
// ---------------------------------------------------------------------------
__global__ __launch_bounds__(1024) void reduce_kernel(
    const float* __restrict__ src, int n, float* __restrict__ dst) {
  __shared__ float red32[32];
  int tid = threadIdx.x;
  float v = 0.0f;
  for (int i = tid; i < n; i += 1024) v += src[i];
  float w = wave_sum32(v);
  if ((tid & 31) == 0) red32[tid >> 5] = w;
  __syncthreads();
  if (tid < 32) {
    float s = wave_sum32(red32[tid]);
    if (tid == 0) dst[0] = s;
  }
}

// ---------------------------------------------------------------------------
// Kernel 7: alpha = rr/pAp on device; x += a*p; r -= a*Ap; partial rr_new.
// ---------------------------------------------------------------------------
__global__ __launch_bounds__(256) void update_xr_kernel(
    const float* __restrict__ p, const float* __restrict__ Ap,
    const float* __restrict__ sc_rr, const float* __restrict__ sc_pAp,
    float* __restrict__ x, float* __restrict__ r,
    float* __restrict__ partials, int n) {
  __shared__ float red8[8];
  int i = blockIdx.x * 256 + threadIdx.x;
  float alpha = sc_rr[0] / sc_pAp[0];
  float acc = 0.0f;
  if (i < n) {
    x[i] += alpha * p[i];
    float rn = r[i] - alpha * Ap[i];
    r[i] = rn;
    acc = rn * rn;
  }
  block_accum256(acc, red8, &partials[blockIdx.x], threadIdx.x);
}

// ---------------------------------------------------------------------------
// Kernel 8: beta = rr_new/rr_old on device; p = r + beta*p.
// ---------------------------------------------------------------------------
__global__ __launch_bounds__(256) void update_p_kernel(
    const float* __restrict__ r, const float* __restrict__ sc_new,
    const float* __restrict__ sc_old, float* __restrict__ p, int n) {
  int i = blockIdx.x * 256 + threadIdx.x;
  if (i >= n) return;
  float beta = sc_new[0] / sc_old[0];
  p[i] = r[i] + beta * p[i];
}

// ---------------------------------------------------------------------------
// Host driver: everything on `stream`, graph-capture safe, deterministic.
// ---------------------------------------------------------------------------
extern "C" void kernel_launch(void* const* d_in, const int* in_sizes, int n_in,
                              void* d_out, int out_size, void* d_ws, size_t ws_size,
                              hipStream_t stream) {
  (void)n_in; (void)out_size; (void)ws_size;

  const float* S    = (const float*)d_in[0];
  const float* hh   = (const float*)d_in[1];
  const float* rey  = (const float*)d_in[2];
  const float* H    = (const float*)d_in[3];
  const float* bed  = (const float*)d_in[4];
  const float* mw   = (const float*)d_in[5];
  const float* ghf  = (const float*)d_in[6];
  const float* llen = (const float*)d_in[7];
  const float* area = (const float*)d_in[8];
  const int*   dtp  = (const int*)d_in[13];

  const int N = in_sizes[0];     // 1,048,576 nodes
  const int L = in_sizes[2];     // 2,095,104 links
  const int NBLK = (N + 255) / 256;   // 4096 (N is a multiple of 256)

  float* out_S  = (float*)d_out;      // new conduit size  [0, N)
  float* x      = out_S + N;          // new head (CG x)   [N, 2N)
  float* out_Re = out_S + 2 * N;      // Reynolds          [2N, 2N+L)

  float* ws = (float*)d_ws;
  float* T  = ws;  ws += L;           // link transmissivity
  float* ML = ws;  ws += L;           // link melt
  float* F  = ws;  ws += N;           // forcing
  float* b  = ws;  ws += N;           // rhs
  float* r  = ws;  ws += N;
  float* p  = ws;  ws += N;
  float* Ap = ws;  ws += N;
  float* pA = ws;  ws += NBLK;        // partials (p.Ap)
  float* pB = ws;  ws += NBLK;        // partials (r.r)
  float* sc = ws;                     // sc[0]=pAp, sc[1]/sc[2]=rr ping-pong

  dim3 b1(256);
  dim3 blk2(16, 16), grd2(NCg / 16, NRg / 16);

  copy_kernel<<<NBLK, b1, 0, stream>>>(hh, x, N);
  link_kernel<<<(L + 255) / 256, b1, 0, stream>>>(S, hh, rey, llen, T, ML, out_Re, L);
  node_kernel<<<NBLK, b1, 0, stream>>>(S, hh, H, bed, mw, ghf, ML, dtp, F, out_S, N, L);
  matvecT_kernel<<<NBLK, b1, 0, stream>>>(F, T, area, b, N);

  // r0 = b - A^TA x0 ; p0 = r0 ; rr -> sc[1]
  normal_op_kernel<<<grd2, blk2, 0, stream>>>(x, T, area, Ap, pA);
  residual_kernel<<<NBLK, b1, 0, stream>>>(b, Ap, r, p, pB, N);
  reduce_kernel<<<1, 1024, 0, stream>>>(pB, NBLK, sc + 1);

  for (int k = 0; k < 50; ++k) {
    int ra = 1 + (k & 1);          // rr_old slot
    int rb = 1 + ((k + 1) & 1);    // rr_new slot
    normal_op_kernel<<<grd2, blk2, 0, stream>>>(p, T, area, Ap, pA);
    reduce_kernel<<<1, 1024, 0, stream>>>(pA, NBLK, sc + 0);
    update_xr_kernel<<<NBLK, b1, 0, stream>>>(p, Ap, sc + ra, sc + 0, x, r, pB, N);
    reduce_kernel<<<1, 1024, 0, stream>>>(pB, NBLK, sc + rb);
    update_p_kernel<<<NBLK, b1, 0, stream>>>(r, sc + rb, sc + ra, p, N);
  }
  // d_out now holds [new_S | new_head (=x) | Re].
}